// GumbelSparseAttention_43087111913991
// MI455X (gfx1250) — compile-verified
//
#include <hip/hip_runtime.h>
#include <hip/hip_bf16.h>
#include <math.h>

#define L_SEQ 2048
#define E_DIM 1024
#define H_NUM 16
#define D_HEAD 64
#define KSPLIT 4          // K-chunks for the logits GEMM (512 K each)

typedef float v2f __attribute__((ext_vector_type(2)));
typedef float v8f __attribute__((ext_vector_type(8)));

// ---------------------------------------------------------------------------
// Kernel 1: q_mean[h][l] = mean_d q[l, h*64+d]        (Q read once, coalesced)
// ---------------------------------------------------------------------------
__global__ void k_qmean(const float* __restrict__ q, float* __restrict__ qmean) {
    __shared__ float part[256];
    const int l = blockIdx.x;
    const int t = threadIdx.x;
    const float4 x = ((const float4*)(q + (size_t)l * E_DIM))[t];
    part[t] = x.x + x.y + x.z + x.w;
    __syncthreads();
    if (t < H_NUM) {
        float s = 0.f;
#pragma unroll
        for (int i = 0; i < 16; ++i) s += part[t * 16 + i];
        qmean[t * L_SEQ + l] = s * (1.0f / D_HEAD);
    }
}

// ---------------------------------------------------------------------------
// Kernel 2a: partial logits tiles via V_WMMA_F32_16X16X4_F32.
// grid = (128 j-tiles, KSPLIT k-chunks), one wave each -> 512 waves.
// A frag: lane m=lane&15, koff=2*(lane>>4): {A[m,k+koff], A[m,k+koff+1]}
// B frag: B[kk][n] = W[j0+n, k+kk]  -> float2 from row j0+(lane&15)
// C/D:   VGPR r <-> row r + 8*(lane>>4), col lane&15
// ---------------------------------------------------------------------------
__global__ void k_logits_wmma(const float* __restrict__ qmean,
                              const float* __restrict__ W,
                              float* __restrict__ cpart) {
    const int lane = threadIdx.x;           // 0..31
    const int jt   = blockIdx.x;            // j-tile 0..127
    const int sp   = blockIdx.y;            // k-chunk 0..KSPLIT-1
    const int j0   = jt * 16;
    const int k0   = sp * (L_SEQ / KSPLIT);
    const int mn   = lane & 15;
    const int koff = (lane >> 4) << 1;      // 0 or 2
    const float* aptr = qmean + (size_t)mn * L_SEQ + k0 + koff;
    const float* bptr = W + (size_t)(j0 + mn) * L_SEQ + k0 + koff;
    v8f c = {};
#pragma unroll 8
    for (int k = 0; k < L_SEQ / KSPLIT; k += 4) {
        v2f a = *(const v2f*)(aptr + k);
        v2f b = *(const v2f*)(bptr + k);
        c = __builtin_amdgcn_wmma_f32_16x16x4_f32(false, a, false, b,
                                                  (short)0, c, false, false);
    }
    float* ct = cpart + (size_t)(jt * KSPLIT + sp) * 256;   // 16x16 tile
    const int rbase = (lane >> 4) * 8;
#pragma unroll
    for (int r = 0; r < 8; ++r)
        ct[(rbase + r) * 16 + mn] = c[r];
}

// ---------------------------------------------------------------------------
// Kernel 2b: reduce KSPLIT partial tiles + bias -> logits[16, 2048]
// Fixed summation order => bitwise-deterministic across replays.
// ---------------------------------------------------------------------------
__global__ void k_logits_reduce(const float* __restrict__ cpart,
                                const float* __restrict__ bias,
                                float* __restrict__ logits) {
    const int idx = blockIdx.x * 256 + threadIdx.x;   // h*2048 + j
    const int h = idx >> 11;
    const int j = idx & (L_SEQ - 1);
    const int jt = j >> 4, col = j & 15;
    const float* p = cpart + (size_t)(jt * KSPLIT) * 256 + h * 16 + col;
    float s = bias[j];
#pragma unroll
    for (int c = 0; c < KSPLIT; ++c) s += p[c * 256];
    logits[idx] = s;
}

// ---------------------------------------------------------------------------
// Kernel 3: per-head argmax of logits + Gumbel(-log(-log u)); first-index ties
// ---------------------------------------------------------------------------
__global__ void k_argmax(const float* __restrict__ logits,
                         const float* __restrict__ u,
                         int* __restrict__ lstar) {
    __shared__ float rv[256];
    __shared__ int   ri[256];
    const int h = blockIdx.x, t = threadIdx.x;
    float best = -INFINITY; int bidx = 0;
    for (int l = t; l < L_SEQ; l += 256) {
        float g = -logf(-logf(u[h * L_SEQ + l]));
        float z = logits[h * L_SEQ + l] + g;
        if (z > best) { best = z; bidx = l; }   // increasing l: keeps first max
    }
    rv[t] = best; ri[t] = bidx;
    __syncthreads();
    for (int s = 128; s > 0; s >>= 1) {
        if (t < s) {
            float v2 = rv[t + s]; int i2 = ri[t + s];
            if (v2 > rv[t] || (v2 == rv[t] && i2 < ri[t])) { rv[t] = v2; ri[t] = i2; }
        }
        __syncthreads();
    }
    if (t == 0) lstar[h] = ri[0];
}

// ---------------------------------------------------------------------------
// Kernel 4a/4b: column mean of V (the output for every non-selected row)
// ---------------------------------------------------------------------------
__global__ void k_vpart(const float* __restrict__ v, float* __restrict__ vpart) {
    const int e  = blockIdx.x * 256 + threadIdx.x;   // 0..1023
    const int m0 = blockIdx.y * 64;
    float s = 0.f;
#pragma unroll 8
    for (int m = m0; m < m0 + 64; ++m) s += v[(size_t)m * E_DIM + e];
    vpart[blockIdx.y * E_DIM + e] = s;
}

__global__ void k_vmean(const float* __restrict__ vpart, float* __restrict__ vmean) {
    const int e = blockIdx.x * 256 + threadIdx.x;
    float s = 0.f;
#pragma unroll
    for (int c = 0; c < 32; ++c) s += vpart[c * E_DIM + e];
    vmean[e] = s * (1.0f / L_SEQ);
}

// ---------------------------------------------------------------------------
// Kernel 5: one single-query attention row per head:
//   special[h,:] = softmax(q[l*_h] . K^T * d^-0.5) @ V        (block = 1 head)
// ---------------------------------------------------------------------------
__global__ void k_special(const float* __restrict__ q,
                          const float* __restrict__ kmat,
                          const float* __restrict__ v,
                          const int* __restrict__ lstar,
                          float* __restrict__ special) {
    __shared__ float qh[D_HEAD];
    __shared__ float s[L_SEQ];
    __shared__ float red[256];
    __shared__ float part[4 * D_HEAD];
    const int h = blockIdx.x, t = threadIdx.x;
    const int ls = lstar[h];
    if (t < D_HEAD) qh[t] = q[(size_t)ls * E_DIM + h * D_HEAD + t];
    __syncthreads();

    // scores: one wave per key row, lane holds a float2 slice of the 64-dot
    const int wave = t >> 5, lane = t & 31;
    const float scale = 0.125f;                    // 64^-0.5
#pragma unroll 4
    for (int m = wave; m < L_SEQ; m += 8) {
        v2f kv = *(const v2f*)(kmat + (size_t)m * E_DIM + h * D_HEAD + 2 * lane);
        float d = kv.x * qh[2 * lane] + kv.y * qh[2 * lane + 1];
#pragma unroll
        for (int off = 16; off > 0; off >>= 1) d += __shfl_xor(d, off, 32);
        if (lane == 0) s[m] = d * scale;
    }
    __syncthreads();

    // max
    float lm = -INFINITY;
    for (int m = t; m < L_SEQ; m += 256) lm = fmaxf(lm, s[m]);
    red[t] = lm; __syncthreads();
    for (int st = 128; st > 0; st >>= 1) {
        if (t < st) red[t] = fmaxf(red[t], red[t + st]);
        __syncthreads();
    }
    const float mx = red[0];
    __syncthreads();

    // exp + sum
    float lsum = 0.f;
    for (int m = t; m < L_SEQ; m += 256) {
        float p = expf(s[m] - mx); s[m] = p; lsum += p;
    }
    red[t] = lsum; __syncthreads();
    for (int st = 128; st > 0; st >>= 1) {
        if (t < st) red[t] += red[t + st];
        __syncthreads();
    }
    const float denom = red[0];
    __syncthreads();

    // weighted sum over V: 4 m-chunks x 64 dims
    const int dd = t & 63, ch = t >> 6;
    float acc = 0.f;
#pragma unroll 8
    for (int m = ch * 512; m < (ch + 1) * 512; ++m)
        acc += s[m] * v[(size_t)m * E_DIM + h * D_HEAD + dd];
    part[ch * D_HEAD + dd] = acc;
    __syncthreads();
    if (t < D_HEAD) {
        float tot = part[t] + part[D_HEAD + t] + part[2 * D_HEAD + t] + part[3 * D_HEAD + t];
        special[h * D_HEAD + t] = tot / denom;
    }
}

// ---------------------------------------------------------------------------
// Kernel 6: out[l, e] = (l == lstar[e>>6]) ? special[e] : vmean[e]
// ---------------------------------------------------------------------------
__global__ void k_scatter(const float* __restrict__ vmean,
                          const float* __restrict__ special,
                          const int* __restrict__ lstar,
                          float* __restrict__ out) {
    const int idx = blockIdx.x * 256 + threadIdx.x;
    const int l = idx >> 10;
    const int e = idx & (E_DIM - 1);
    const int h = e >> 6;
    out[idx] = (l == lstar[h]) ? special[e] : vmean[e];
}

// ---------------------------------------------------------------------------
extern "C" void kernel_launch(void* const* d_in, const int* in_sizes, int n_in,
                              void* d_out, int out_size, void* d_ws, size_t ws_size,
                              hipStream_t stream) {
    const float* q    = (const float*)d_in[0];   // [2048,1024]
    const float* kmat = (const float*)d_in[1];   // [2048,1024]
    const float* v    = (const float*)d_in[2];   // [2048,1024]
    const float* W    = (const float*)d_in[3];   // [2048,2048]
    const float* bias = (const float*)d_in[4];   // [2048]
    const float* u    = (const float*)d_in[5];   // [16,2048]
    float* out = (float*)d_out;

    float* ws      = (float*)d_ws;
    float* qmean   = ws;                   // 16*2048            = 32768
    float* cpart   = ws + 32768;           // 128*KSPLIT*256     = 131072
    float* logits  = ws + 163840;          // 16*2048            = 32768
    float* vpart   = ws + 196608;          // 32*1024            = 32768
    float* vmean   = ws + 229376;          // 1024
    float* special = ws + 230400;          // 16*64              = 1024
    int*   lstar   = (int*)(ws + 231424);  // 16

    k_qmean        <<<L_SEQ, 256, 0, stream>>>(q, qmean);
    k_logits_wmma  <<<dim3(L_SEQ / 16, KSPLIT), 32, 0, stream>>>(qmean, W, cpart);
    k_logits_reduce<<<(H_NUM * L_SEQ) / 256, 256, 0, stream>>>(cpart, bias, logits);
    k_argmax       <<<H_NUM, 256, 0, stream>>>(logits, u, lstar);
    k_vpart        <<<dim3(E_DIM / 256, 32), 256, 0, stream>>>(v, vpart);
    k_vmean        <<<E_DIM / 256, 256, 0, stream>>>(vpart, vmean);
    k_special      <<<H_NUM, 256, 0, stream>>>(q, kmat, v, lstar, special);
    k_scatter      <<<(L_SEQ * E_DIM) / 256, 256, 0, stream>>>(vmean, special, lstar, out);
}